// FasterRCNN_67456756351306
// MI455X (gfx1250) — compile-verified
//
#include <hip/hip_runtime.h>
#include <hip/hip_bf16.h>
#include <cstdint>

// Problem constants (match reference)
#define BB    8
#define NN    65536
#define MM    64
#define IMGF  500.0f
#define MINS  10.0f

typedef __attribute__((ext_vector_type(4))) unsigned int uint32x4_t_;
typedef __attribute__((ext_vector_type(8))) int          int32x8_t_;
typedef __attribute__((ext_vector_type(4))) int          int32x4_t_;

// One block = 256 proposals of one batch. gt tile (64 boxes, 1 KB) staged into
// LDS once per block via the Tensor Data Mover (CDNA5 TDM, TENSORcnt-tracked).
__global__ __launch_bounds__(256) void rcnn_targets_kernel(
    const float4* __restrict__ boxes,   // [B*N]
    const float4* __restrict__ deltas,  // [B*N]
    const float4* __restrict__ gt,      // [B*M]
    float4* __restrict__ out)           // [B*N]
{
    __shared__ float4 s_gt[MM];
    __shared__ float  s_area[MM];

    const int blocks_per_batch = NN / 256;
    const int b   = blockIdx.x / blocks_per_batch;
    const int n   = (blockIdx.x % blocks_per_batch) * 256 + threadIdx.x;
    const int tid = threadIdx.x;

#if defined(__gfx1250__) && __has_builtin(__builtin_amdgcn_tensor_load_to_lds)
    // --- TDM: DMA the 64x4 f32 gt tile for this batch into LDS -------------
    if (tid < 32) {  // whole wave 0 active (TDM issues per-wave; no execz skip)
        const uint64_t ga = (uint64_t)(uintptr_t)(gt + (size_t)b * MM);
        const uint32_t la = (uint32_t)(uintptr_t)(void*)&s_gt[0]; // LDS offset = low 32b of flat addr

        uint32x4_t_ g0;
        g0[0] = 1u;                                            // count=1, user D#
        g0[1] = la;                                            // lds_addr [63:32]
        g0[2] = (uint32_t)ga;                                  // global_addr[31:0]
        g0[3] = ((uint32_t)(ga >> 32) & 0x01FFFFFFu)           // global_addr[56:32]
                | 0x80000000u;                                 // type=2 ("image") [127:126]

        int32x8_t_ g1;
        g1[0] = 0x00020000;                  // data_size=2 (4 B); no multicast/pad/iterate
        g1[1] = (int)((MM * 4) << 16);       // tensor_dim0[15:0]=256 elems  (bits 63:48)
        g1[2] = (int)(1u << 16);             // tensor_dim0[31:16]=0, tensor_dim1=1
        g1[3] = (int)((MM * 4) << 16);       // tile_dim0=256 elems          (bits 127:112)
        g1[4] = 1;                           // tile_dim1=1, tile_dim2=0
        g1[5] = MM * 4;                      // tensor_dim0_stride[31:0]=256
        g1[6] = (int)((MM * 4) << 16);       // stride0[47:32]=0, tensor_dim1_stride lo=256
        g1[7] = 0;

        int32x4_t_ z4 = {0, 0, 0, 0};        // groups 2/3 unused (<=2D tensor)
        int32x8_t_ z8 = {0, 0, 0, 0, 0, 0, 0, 0};
        __builtin_amdgcn_tensor_load_to_lds(g0, g1, z4, z4, z8, 0);
        __builtin_amdgcn_s_wait_tensorcnt(0);
    }
    __syncthreads();
#else
    if (tid < MM) s_gt[tid] = gt[(size_t)b * MM + tid];
    __syncthreads();
#endif

    // Precompute gt areas (torchvision convention: no +1)
    if (tid < MM) {
        const float4 g = s_gt[tid];
        s_area[tid] = (g.z - g.x) * (g.w - g.y);
    }
    __syncthreads();

    // ---- decode proposal (bbox2proposal, +1 convention) -------------------
    const size_t idx = (size_t)b * NN + n;
    const float4 bx = boxes[idx];
    const float4 dl = deltas[idx];

    const float w  = bx.z - bx.x + 1.0f;
    const float h  = bx.w - bx.y + 1.0f;
    const float cx = bx.x + 0.5f * w;
    const float cy = bx.y + 0.5f * h;
    const float pcx = dl.x * w + cx;
    const float pcy = dl.y * h + cy;
    const float pw  = __expf(dl.z) * w;
    const float ph  = __expf(dl.w) * h;

    // max(.,0) then clip(0,IMG) == clamp to [0, IMG]
    const float x1 = fminf(fmaxf(pcx - 0.5f * pw, 0.0f), IMGF);
    const float y1 = fminf(fmaxf(pcy - 0.5f * ph, 0.0f), IMGF);
    const float x2 = fminf(fmaxf(pcx + 0.5f * pw, 0.0f), IMGF);
    const float y2 = fminf(fmaxf(pcy + 0.5f * ph, 0.0f), IMGF);

    const float dw = x2 - x1;
    const float dh = y2 - y1;
    const bool keep = (dw >= MINS) && (dh >= MINS);
    const float area_d = dw * dh;

    // ---- IoU max / argmax over 64 gt boxes (LDS broadcast reads) ----------
    float best = -1.0f;
    int   bidx = 0;
#pragma unroll 8
    for (int m = 0; m < MM; ++m) {
        const float4 g  = s_gt[m];
        const float iw  = fmaxf(fminf(g.z, x2) - fmaxf(g.x, x1), 0.0f);
        const float ih  = fmaxf(fminf(g.w, y2) - fmaxf(g.y, y1), 0.0f);
        const float inter = iw * ih;
        const float uni   = s_area[m] + area_d - inter;   // > 0 always (gt area > 0)
        const float iou   = inter * __builtin_amdgcn_rcpf(uni);
        if (iou > best) { best = iou; bidx = m; }         // strict '>' => first argmax
    }

    // ---- regression targets vs matched gt (proposal2deltas) ---------------
    const float ex_w  = dw + 1.0f;
    const float ex_h  = dh + 1.0f;
    const float ex_cx = x1 + 0.5f * ex_w;
    const float ex_cy = y1 + 0.5f * ex_h;

    const float4 g  = s_gt[bidx];
    const float gw  = g.z - g.x + 1.0f;
    const float gh  = g.w - g.y + 1.0f;
    const float gcx = g.x + 0.5f * gw;
    const float gcy = g.y + 0.5f * gh;

    const float msk = (best > 0.5f && keep) ? 1.0f : 0.0f;

    float4 r;
    r.x = msk * ((gcx - ex_cx) / ex_w);
    r.y = msk * ((gcy - ex_cy) / ex_h);
    r.z = msk * __logf(gw / ex_w);
    r.w = msk * __logf(gh / ex_h);

    out[idx] = r;
}

extern "C" void kernel_launch(void* const* d_in, const int* in_sizes, int n_in,
                              void* d_out, int out_size, void* d_ws, size_t ws_size,
                              hipStream_t stream) {
    (void)in_sizes; (void)n_in; (void)out_size; (void)d_ws; (void)ws_size;
    const float4* boxes  = (const float4*)d_in[0];  // [B,N,4] f32
    const float4* deltas = (const float4*)d_in[1];  // [B,N,4] f32
    const float4* gt     = (const float4*)d_in[2];  // [B,M,4] f32
    float4*       out    = (float4*)d_out;          // [B,N,4] f32

    const dim3 grid(BB * (NN / 256));
    const dim3 block(256);
    rcnn_targets_kernel<<<grid, block, 0, stream>>>(boxes, deltas, gt, out);
}